// SparseHead1_8839042695387
// MI455X (gfx1250) — compile-verified
//
#include <hip/hip_runtime.h>
#include <hip/hip_bf16.h>

// Reference: coords for causal-in-block sparse attention, k=8, t=8192.
//   pairs_per_block = k*(k+1)/2 = 36
//   out[(b*36 + p)*2 + 0] = tril_row(p) + b*k
//   out[(b*36 + p)*2 + 1] = tril_col(p) + b*k
// Output: 36864 pairs = 73728 f32 = 288 KB. Write-only, launch-latency bound
// on MI455X (288 KB / 23.3 TB/s ~ 13 ns). No matrix math exists in this
// workload; the optimization targets are zero input reads, register-only
// coordinate arithmetic, and 16B non-temporal stores.

#define KBLK 8
#define PAIRS_PER_BLK 36   // KBLK*(KBLK+1)/2

// Native clang vector type: required by __builtin_nontemporal_store
// (HIP_vector_type float4 is a class and is rejected by the builtin).
typedef __attribute__((ext_vector_type(4))) float v4f;

__device__ __forceinline__ void pair_coords(int p, float& r, float& c) {
    // p = global pair index; blk = p/36 lowers to umulhi magic multiply.
    int blk = p / PAIRS_PER_BLK;
    int idx = p - blk * PAIRS_PER_BLK;
    // Row = count of triangular numbers <= idx (tri(1..7) = 1,3,6,10,15,21,28).
    int row = (idx >= 1) + (idx >= 3) + (idx >= 6) + (idx >= 10) +
              (idx >= 15) + (idx >= 21) + (idx >= 28);
    int col = idx - ((row * (row + 1)) >> 1);
    float off = (float)(blk * KBLK);
    r = (float)row + off;
    c = (float)col + off;
}

__global__ __launch_bounds__(256) void coords_b128_kernel(v4f* __restrict__ out,
                                                          int n_vec) {
    int i = blockIdx.x * blockDim.x + threadIdx.x;
    if (i >= n_vec) return;
    float r0, c0, r1, c1;
    pair_coords(2 * i,     r0, c0);   // pair 2i   -> (row, col)
    pair_coords(2 * i + 1, r1, c1);   // pair 2i+1 -> (row, col)
    v4f v = { r0, c0, r1, c1 };
    // Streaming, write-once output: NT temporal hint, 16B global_store_b128.
    __builtin_nontemporal_store(v, &out[i]);
}

extern "C" void kernel_launch(void* const* d_in, const int* in_sizes, int n_in,
                              void* d_out, int out_size, void* d_ws, size_t ws_size,
                              hipStream_t stream) {
    (void)d_in; (void)in_sizes; (void)n_in; (void)d_ws; (void)ws_size;
    // out_size = t*(k+1)/2 * 2 floats. Each block contributes 72 floats
    // (divisible by 4), so out_size is always a multiple of 4: no tail.
    v4f* out = (v4f*)d_out;
    int n_vec = out_size >> 2;                 // v4f stores (2 pairs each)
    int threads = 256;                         // 8 wave32s per workgroup
    int blocks = (n_vec + threads - 1) / threads;
    if (blocks < 1) blocks = 1;
    coords_b128_kernel<<<blocks, threads, 0, stream>>>(out, n_vec);
}